// DotAttention_19765439497049
// MI455X (gfx1250) — compile-verified
//
#include <hip/hip_runtime.h>
#include <math.h>

typedef __attribute__((ext_vector_type(2))) float v2f;
typedef __attribute__((ext_vector_type(8))) float v8f;

#define NB 4
#define NQ 4096
#define NW 5
#define NS 64
#define ND 512
#define QT 128                 // q-rows per block (8 waves x 16 rows)
#define KSTRIDE 516            // padded LDS row stride for K tile (floats)
#define SSTRIDE 68             // padded LDS row stride for s-hat staging (floats)
#define SH_FLOATS (NS*KSTRIDE + NS + 8*16*SSTRIDE)

__global__ __launch_bounds__(256) void cosattn_wmma(
    const float* __restrict__ Qm,   // [B, Q, D]
    const float* __restrict__ Km,   // [B, W, S, D]
    float* __restrict__ Om)         // [B, Q, W, D]
{
  extern __shared__ __align__(16) float lds[];
  float* shK    = lds;                      // NS * KSTRIDE
  float* shKinv = shK + NS * KSTRIDE;       // NS
  float* shS    = shKinv + NS;              // 8 * 16 * SSTRIDE

  const int tid  = threadIdx.x;
  const int lane = tid & 31;
  const int wv   = tid >> 5;                // wave id 0..7
  const int half = lane >> 4;               // 0 | 1
  const int l16  = lane & 15;

  int blk = blockIdx.x;
  const int qt = blk & 31;                  // Q/QT == 32
  blk >>= 5;
  const int w = blk % NW;
  const int b = blk / NW;

  const float* Kbase = Km + (size_t)(b * NW + w) * NS * ND;
  const float* Qbase = Qm + ((size_t)b * NQ + (size_t)qt * QT) * ND;
  float*       Obase = Om + (((size_t)b * NQ + (size_t)qt * QT) * NW + w) * ND;

  // ------- Stage 1a: K tile -> LDS via CDNA5 async global->LDS DMA ---------
  // Wave wv owns rows s = wv*8 .. wv*8+7. Each lane issues a 16B async copy
  // (ASYNCcnt-tracked, no VGPR staging); dynamic LDS starts at offset 0.
  for (int r = 0; r < 8; ++r) {
    const int s = wv * 8 + r;
    const float* gk = Kbase + (size_t)s * ND;
    const unsigned lbase = (unsigned)(s * KSTRIDE) * 4u;  // LDS byte offset
#pragma unroll
    for (int it = 0; it < 4; ++it) {
      const unsigned loff = lbase + (unsigned)((it * 32 + lane) * 16);
      const float* g = gk + (it * 32 + lane) * 4;
      asm volatile("global_load_async_to_lds_b128 %0, %1, off"
                   :: "v"(loff), "v"(g) : "memory");
    }
  }
  asm volatile("s_wait_asynccnt 0x0" ::: "memory");
  __syncthreads();

  // ------- Stage 1b: per-row 1/||k_s|| from the LDS-resident tile ----------
  for (int r = 0; r < 8; ++r) {
    const int s = wv * 8 + r;
    const float* lrow = shK + s * KSTRIDE;
    float ss = 0.f;
#pragma unroll
    for (int it = 0; it < 4; ++it) {
      const int d = (it * 32 + lane) * 4;
      const float4 v = *reinterpret_cast<const float4*>(lrow + d);
      ss += v.x * v.x + v.y * v.y + v.z * v.z + v.w * v.w;
    }
    ss += __shfl_xor(ss, 1);
    ss += __shfl_xor(ss, 2);
    ss += __shfl_xor(ss, 4);
    ss += __shfl_xor(ss, 8);
    ss += __shfl_xor(ss, 16);
    if (lane == 0) shKinv[s] = 1.f / fmaxf(sqrtf(ss), 1e-12f);
  }
  __syncthreads();

  // ---------------- Stage 2: GEMM1  t[16q x 64s] = Qtile x K^T --------------
  // V_WMMA_F32_16X16X4_F32: A 16x4 (lane: M=l16, K=2*half+{0,1}),
  //                         B 4x16 (lane: N=l16, K=2*half+{0,1}),
  //                         C/D 16x16 (lane: N=l16, VGPR v: M=v+8*half).
  v8f zero = {};
  v8f acc[4];
  acc[0] = zero; acc[1] = zero; acc[2] = zero; acc[3] = zero;

  const float* qrow = Qbase + (size_t)(wv * 16 + l16) * ND + 2 * half;
  const float* kcol = shK + 2 * half;
#pragma unroll 2
  for (int k = 0; k < ND; k += 4) {
    const v2f a = *reinterpret_cast<const v2f*>(qrow + k);
#pragma unroll
    for (int j = 0; j < 4; ++j) {
      const v2f bb = *reinterpret_cast<const v2f*>(kcol + (j * 16 + l16) * KSTRIDE + k);
      acc[j] = __builtin_amdgcn_wmma_f32_16x16x4_f32(
          false, a, false, bb, (short)0, acc[j], false, false);
    }
  }

  // ------- Stage 3: fold kinv, l2-normalize each q-row over s, stage to LDS --
#pragma unroll
  for (int j = 0; j < 4; ++j) {
    const float ks = shKinv[j * 16 + l16];   // per-lane column scale (s index)
    acc[j] *= ks;
  }
  float rs[8];
#pragma unroll
  for (int v = 0; v < 8; ++v) {
    float p = acc[0][v] * acc[0][v] + acc[1][v] * acc[1][v]
            + acc[2][v] * acc[2][v] + acc[3][v] * acc[3][v];
    // reduce over the 16 lanes of this half (all hold the same row M=v+8*half)
    p += __shfl_xor(p, 1);
    p += __shfl_xor(p, 2);
    p += __shfl_xor(p, 4);
    p += __shfl_xor(p, 8);
    rs[v] = 1.f / fmaxf(sqrtf(p), 1e-12f);
  }
  float* shSw = shS + wv * 16 * SSTRIDE;     // this wave's 16x64 s-hat tile
#pragma unroll
  for (int j = 0; j < 4; ++j) {
#pragma unroll
    for (int v = 0; v < 8; ++v) {
      shSw[(v + 8 * half) * SSTRIDE + j * 16 + l16] = acc[j][v] * rs[v];
    }
  }
  __syncthreads();

  // ---------------- Stage 4: GEMM2  out[16q x 512d] = s-hat x K -------------
  const float* arow = shSw + l16 * SSTRIDE + 2 * half;
#pragma unroll 1
  for (int nc = 0; nc < 8; ++nc) {           // 8 chunks of 64 output columns
    v8f o[4];
    o[0] = zero; o[1] = zero; o[2] = zero; o[3] = zero;
#pragma unroll 2
    for (int ks = 0; ks < NS; ks += 4) {
      const v2f a = *reinterpret_cast<const v2f*>(arow + ks);
      const float* kr0 = shK + (ks + 2 * half) * KSTRIDE + nc * 64 + l16;
#pragma unroll
      for (int j = 0; j < 4; ++j) {
        v2f bb;
        bb.x = kr0[j * 16];                  // K[s=ks+2*half  ][d]
        bb.y = kr0[j * 16 + KSTRIDE];        // K[s=ks+2*half+1][d]
        o[j] = __builtin_amdgcn_wmma_f32_16x16x4_f32(
            false, a, false, bb, (short)0, o[j], false, false);
      }
    }
    // Streamed output: non-temporal stores so the 168 MB result doesn't evict
    // the 5x-reused query tiles from L2.
#pragma unroll
    for (int j = 0; j < 4; ++j) {
#pragma unroll
      for (int v = 0; v < 8; ++v) {
        float* optr = Obase
            + (size_t)(wv * 16 + v + 8 * half) * (NW * ND)
            + nc * 64 + j * 16 + l16;
        __builtin_nontemporal_store(o[j][v], optr);
      }
    }
  }
}

extern "C" void kernel_launch(void* const* d_in, const int* in_sizes, int n_in,
                              void* d_out, int out_size, void* d_ws, size_t ws_size,
                              hipStream_t stream) {
  (void)in_sizes; (void)n_in; (void)out_size; (void)d_ws; (void)ws_size;
  const float* q = (const float*)d_in[0];
  const float* k = (const float*)d_in[1];
  float*       o = (float*)d_out;

  const size_t shbytes = (size_t)SH_FLOATS * sizeof(float);   // ~163 KB
  (void)hipFuncSetAttribute(reinterpret_cast<const void*>(cosattn_wmma),
                            hipFuncAttributeMaxDynamicSharedMemorySize,
                            (int)shbytes);

  dim3 grid(NB * NW * (NQ / QT));   // 640 blocks
  cosattn_wmma<<<grid, 256, shbytes, stream>>>(q, k, o);
}